// TargetEncoder_532575944857
// MI455X (gfx1250) — compile-verified
//
#include <hip/hip_runtime.h>
#include <hip/hip_bf16.h>
#include <stdint.h>

// TargetEncoder scatter-max: out[b, v] = max over s of (ids[b,s]==v ? w[b,s] : 0), bf16.
// B=1024, S=256, V=30522. Pure store-bandwidth problem (~62.5 MB bf16 out).
// Strategy: per-row scatter-max in LDS (ds_max_u32 on f32 bit patterns),
// convert to packed bf16 in LDS, stream to HBM with CDNA5 async
// global_store_async_from_lds_b32 (ASYNCcnt), 3 vocab chunks per row.

#define S_DIM    256
#define V_DIM    30522
#define NCHUNK   3
#define VCHUNK   (V_DIM / NCHUNK)   // 10174 f32 slots per chunk
#define PKCHUNK  (VCHUNK / 2)       // 5087 packed (2x bf16) words per chunk
#define NTHREADS 1024               // 32 wave32 per workgroup
#define D_ITERS  ((PKCHUNK + NTHREADS - 1) / NTHREADS)  // 5 uniform iterations

__device__ __forceinline__ unsigned bf16_rne_bits(unsigned u) {
  // round-to-nearest-even f32 -> bf16 (inputs here are finite, >= 0)
  return u + 0x7FFFu + ((u >> 16) & 1u);
}

__global__ __launch_bounds__(NTHREADS)
void target_encoder_scatter_max(const int* __restrict__ ids,
                                const float* __restrict__ wts,
                                __hip_bfloat16* __restrict__ out)
{
  __shared__ unsigned sm_val[VCHUNK + 2];  // f32 bit patterns (scatter-max accumulators)
  __shared__ unsigned sm_pk[PKCHUNK + 1];  // packed bf16 pairs (async-store staging)

  const int b   = blockIdx.x;
  const int tid = threadIdx.x;

  // Each of the first 256 threads owns one token of this row.
  int      my_id = 0;
  unsigned my_w  = 0;
  if (tid < S_DIM) {
    my_id = ids[b * S_DIM + tid];
    my_w  = __float_as_uint(wts[b * S_DIM + tid]);  // uniform [0,1): bits order as u32
  }

  const unsigned long long row_base =
      (unsigned long long)(uintptr_t)out +
      (unsigned long long)b * (unsigned long long)(V_DIM * 2);

  for (int c = 0; c < NCHUNK; ++c) {
    const int vlo = c * VCHUNK;

    // Phase A: zero the scatter region (baseline 0 of the max).
    for (int i = tid; i < VCHUNK; i += NTHREADS) sm_val[i] = 0u;
    __syncthreads();

    // Phase B: LDS scatter-max (ds_max_u32).
    if (tid < S_DIM) {
      int local = my_id - vlo;
      if (local >= 0 && local < VCHUNK) atomicMax(&sm_val[local], my_w);
    }
    __syncthreads();

    // Before overwriting this thread's staging words, drain this thread's
    // async stores from the previous chunk (it issued the stores of exactly
    // the sm_pk words it writes below).
    asm volatile("s_wait_asynccnt 0" ::: "memory");

    // Phase C: f32 -> packed bf16 (RNE) into staging.
    for (int i = tid; i < PKCHUNK; i += NTHREADS) {
      unsigned u0 = bf16_rne_bits(sm_val[2 * i]);
      unsigned u1 = bf16_rne_bits(sm_val[2 * i + 1]);
      sm_pk[i] = (u0 >> 16) | (u1 & 0xFFFF0000u);
    }
    __syncthreads();

    // Phase D: async-stream LDS -> HBM, 4 B per lane per op. Uniform trip
    // count so every lane of every wave is active (async ops and EXEC
    // masking are conservative); tail lanes clamp to a word this same
    // thread already stored (same data, same address -> benign duplicate).
    const unsigned long long chunk_base =
        row_base + (unsigned long long)(vlo * 2);
    for (int k = 0; k < D_ITERS; ++k) {
      int i = tid + k * NTHREADS;
      int j = (i < PKCHUNK) ? i : (i - NTHREADS);  // clamp to own prior word
      unsigned lds_off = (unsigned)(uintptr_t)&sm_pk[j];  // low 32 bits = LDS byte addr
      unsigned long long gaddr = chunk_base + 4ull * (unsigned)j;
      asm volatile("global_store_async_from_lds_b32 %0, %1, off"
                   :: "v"(gaddr), "v"(lds_off)
                   : "memory");
    }
    __syncthreads();  // all threads done reading sm_val before next zero pass
  }

  // Drain remaining async stores (S_ENDPGM also waits; be explicit).
  asm volatile("s_wait_asynccnt 0" ::: "memory");
}

extern "C" void kernel_launch(void* const* d_in, const int* in_sizes, int n_in,
                              void* d_out, int out_size, void* d_ws, size_t ws_size,
                              hipStream_t stream) {
  const int*      ids = (const int*)d_in[0];     // (B, S) int32
  const float*    w   = (const float*)d_in[1];   // (B, S, 1) f32
  __hip_bfloat16* out = (__hip_bfloat16*)d_out;  // (B, V) bf16

  const int B = in_sizes[0] / S_DIM;  // 1024
  target_encoder_scatter_max<<<dim3(B), dim3(NTHREADS), 0, stream>>>(ids, w, out);

  (void)n_in; (void)out_size; (void)d_ws; (void)ws_size;
}